// MultiGraphFilter_90701119357382
// MI455X (gfx1250) — compile-verified
//
#include <hip/hip_runtime.h>

// Problem constants (from the reference)
#define B_    4
#define N_    8192
#define G_    8
#define F_    8
#define NUMS_ 3
#define K_    4            // NUM_S + 1
#define GN    (G_ * N_)    // 65536
#define KGN   (K_ * GN)    // 262144 floats per batch in z
#define MCHUNKS 8
#define MCHUNK  (N_ / MCHUNKS)   // 1024

typedef __attribute__((ext_vector_type(2))) float v2f;
typedef __attribute__((ext_vector_type(8))) float v8f;

// ---------------------------------------------------------------------------
// Kernel 1: pack x[B,N,G] -> z[b,0,g,n] and zero z[b,1..3,:,:] (accumulators).
// ---------------------------------------------------------------------------
__global__ __launch_bounds__(256) void pack_kernel(const float* __restrict__ x,
                                                   float* __restrict__ z) {
    int idx = blockIdx.x * blockDim.x + threadIdx.x;   // over B*G*N
    if (idx >= B_ * G_ * N_) return;
    int n = idx % N_;
    int g = (idx / N_) % G_;
    int b = idx / GN;
    size_t base = (size_t)b * KGN + (size_t)g * N_ + n;
    z[base + 0 * (size_t)GN] = x[((size_t)b * N_ + n) * G_ + g];
    z[base + 1 * (size_t)GN] = 0.0f;
    z[base + 2 * (size_t)GN] = 0.0f;
    z[base + 3 * (size_t)GN] = 0.0f;
}

// ---------------------------------------------------------------------------
// Kernel 2: one propagation step  z[:,kout,g,n] += sum_m S[n,m] * z[:,kin,g,m]
// D[16x16] tiles via V_WMMA_F32_16X16X4_F32.
//   wave -> (row tile n0 = tile*16, m-chunk of 1024)
//   A lane layout (f32 16x4): lanes 0-15 hold K=0,1 ; lanes 16-31 hold K=2,3
//   B lane layout (f32 4x16): VGPR j, lane l -> (K = 2*(l>>4)+j, col = l&15)
//   D: VGPR v, lanes 0-15 -> row n0+v ; lanes 16-31 -> row n0+v+8
// Partial sums over m-chunks reduced with global_atomic_add_f32.
// ---------------------------------------------------------------------------
__global__ __launch_bounds__(256) void wmma_step_kernel(const float* __restrict__ S,
                                                        float* __restrict__ z,
                                                        int kin, int kout) {
    const int lane = threadIdx.x & 31;
    const int wave = threadIdx.x >> 5;
    const int wid  = blockIdx.x * (blockDim.x >> 5) + wave;  // 0 .. 4095
    const int tile  = wid >> 3;          // 0 .. 511  (16 output rows each)
    const int chunk = wid & (MCHUNKS-1); // 0 .. 7    (m reduction slice)
    const int h  = lane >> 4;            // half-wave select
    const int M  = lane & 15;
    const int n0 = tile * 16;
    const int m0 = chunk * MCHUNK;

    // A stream: 16 rows of S, this lane covers K-offsets {2h, 2h+1}
    const float* __restrict__ Arow = S + (size_t)(n0 + M) * N_ + (size_t)m0 + 2 * h;

    // B streams: columns c = cb*16 + M ; c -> (b = c>>3, g = c&7)
    const int c0 = M, c1 = 16 + M;
    const float* __restrict__ Bp0 = z + (size_t)(c0 >> 3) * KGN + (size_t)kin * GN
                                      + (size_t)(c0 & 7) * N_ + (size_t)m0 + 2 * h;
    const float* __restrict__ Bp1 = z + (size_t)(c1 >> 3) * KGN + (size_t)kin * GN
                                      + (size_t)(c1 & 7) * N_ + (size_t)m0 + 2 * h;

    v8f acc0 = {};
    v8f acc1 = {};
    for (int m = 0; m < MCHUNK; m += 8) {
        __builtin_prefetch(Arow + m + 512, 0, 3);       // near-cache prefetch of S stream
        v2f a0  = *(const v2f*)(Arow + m);
        v2f a1  = *(const v2f*)(Arow + m + 4);
        v2f b00 = *(const v2f*)(Bp0 + m);
        v2f b01 = *(const v2f*)(Bp0 + m + 4);
        v2f b10 = *(const v2f*)(Bp1 + m);
        v2f b11 = *(const v2f*)(Bp1 + m + 4);
        // interleave accumulators to avoid back-to-back D->C chains
        acc0 = __builtin_amdgcn_wmma_f32_16x16x4_f32(false, a0, false, b00, (short)0, acc0, false, false);
        acc1 = __builtin_amdgcn_wmma_f32_16x16x4_f32(false, a0, false, b10, (short)0, acc1, false, false);
        acc0 = __builtin_amdgcn_wmma_f32_16x16x4_f32(false, a1, false, b01, (short)0, acc0, false, false);
        acc1 = __builtin_amdgcn_wmma_f32_16x16x4_f32(false, a1, false, b11, (short)0, acc1, false, false);
    }

    float* __restrict__ Op0 = z + (size_t)(c0 >> 3) * KGN + (size_t)kout * GN + (size_t)(c0 & 7) * N_;
    float* __restrict__ Op1 = z + (size_t)(c1 >> 3) * KGN + (size_t)kout * GN + (size_t)(c1 & 7) * N_;
#pragma unroll
    for (int v = 0; v < 8; ++v) {
        int row = n0 + v + 8 * h;
        atomicAdd(Op0 + row, acc0[v]);
        atomicAdd(Op1 + row, acc1[v]);
    }
}

// ---------------------------------------------------------------------------
// Kernel 3: scrambled-reshape epilogue (faithful to the torch raw reshapes):
//   y[b,i,f] = sum_{jj<32} z[b].flat[i*32+jj] * w.flat[jj*8+f]
//   out[b].flat[m] = y[b].flat[m] + bias[m / N]
// One thread per (b,i): 32 contiguous z floats in, 8 contiguous outs.
// ---------------------------------------------------------------------------
__global__ __launch_bounds__(256) void final_kernel(const float* __restrict__ z,
                                                    const float* __restrict__ w,
                                                    const float* __restrict__ bias,
                                                    float* __restrict__ out) {
    int t = blockIdx.x * blockDim.x + threadIdx.x;     // over B*N
    if (t >= B_ * N_) return;
    int b = t / N_;
    int i = t % N_;
    const float* __restrict__ zp = z + (size_t)b * KGN + (size_t)i * 32;

    float acc[F_];
#pragma unroll
    for (int j = 0; j < F_; ++j) acc[j] = 0.0f;
#pragma unroll
    for (int jj = 0; jj < 32; ++jj) {
        float zv = zp[jj];
#pragma unroll
        for (int j = 0; j < F_; ++j) acc[j] += zv * w[jj * F_ + j];   // uniform -> scalar loads
    }

    float* __restrict__ op = out + (size_t)b * (N_ * F_) + (size_t)i * F_;
#pragma unroll
    for (int j = 0; j < F_; ++j) {
        int mflat = i * F_ + j;
        op[j] = acc[j] + bias[mflat >> 13];            // mflat / N_
    }
}

// ---------------------------------------------------------------------------
extern "C" void kernel_launch(void* const* d_in, const int* in_sizes, int n_in,
                              void* d_out, int out_size, void* d_ws, size_t ws_size,
                              hipStream_t stream) {
    const float* x    = (const float*)d_in[0];   // [B,N,G]
    const float* S    = (const float*)d_in[1];   // [NUM_S,N,N]
    const float* w    = (const float*)d_in[2];   // [F,1,K,G] raw
    const float* bias = (const float*)d_in[3];   // [F,1]
    float* out = (float*)d_out;
    float* z   = (float*)d_ws;                   // needs B*K*G*N*4 = 4 MB

    // 1) transpose-pack x into z[:,0] and zero the accumulation slices
    pack_kernel<<<(B_ * G_ * N_ + 255) / 256, 256, 0, stream>>>(x, z);

    // 2) three chained skinny GEMMs, each streaming one 256 MB S matrix once
    const int nblocks = (512 * MCHUNKS) / 8;     // 4096 waves, 8 waves/block
    for (int k = 1; k <= NUMS_; ++k) {
        wmma_step_kernel<<<nblocks, 256, 0, stream>>>(S + (size_t)(k - 1) * N_ * N_, z, k - 1, k);
    }

    // 3) epilogue: scrambled 32->8 projection + bias
    final_kernel<<<(B_ * N_ + 255) / 256, 256, 0, stream>>>(z, w, bias, out);
}